// Net_16647293239829
// MI455X (gfx1250) — compile-verified
//
#include <hip/hip_runtime.h>

// Problem sizes (fixed by the reference)
#define BQ    512
#define NPTS  8192
#define DIN   128
#define DMID  256
#define DOUT  16

typedef __attribute__((ext_vector_type(2))) float v2f;
typedef __attribute__((ext_vector_type(8))) float v8f;

// ---------------------------------------------------------------------------
// MLP: out[r, 0:16] = relu(X[r, :] @ W1^T) @ W2^T
// One wave (32 lanes) computes a 16-row tile using V_WMMA_F32_16X16X4_F32.
// 4 waves per block, each on its own tile; no inter-wave sharing.
//
// Fragment layouts (ISA 7.12.2, wave32):
//   A 16x4 f32 : lane L -> row M = L&15 ; v2f holds K = k0 + 2*(L>>4) + {0,1}
//   B 4x16 f32 : lane L -> col N = L&15 ; v2f holds K = k0 + 2*(L>>4) + {0,1}
//   C/D 16x16  : VGPR r, lane L -> (M = r + 8*(L>>4), N = L&15)
//
// The whole 16x128 A operand is 32 v2f per lane -> held in VGPRs for all 16
// mid-tiles (no LDS staging, no DS waits in the fc1 WMMA stream).
// ---------------------------------------------------------------------------
__global__ __launch_bounds__(128) void mlp_wmma(const float* __restrict__ X,
                                                const float* __restrict__ W1,
                                                const float* __restrict__ W2,
                                                float* __restrict__ out) {
    __shared__ __align__(16) float hs[4][16][18];   // relu(H) 16x16 tile per wave

    const int w       = threadIdx.x >> 5;   // wave in block
    const int lane    = threadIdx.x & 31;
    const int col     = lane & 15;          // N (and A-row M)
    const int hi      = lane >> 4;          // 0: lanes 0-15, 1: lanes 16-31
    const int koff    = 2 * hi;
    const int tile    = blockIdx.x * 4 + w;
    const int rowbase = tile * 16;

    // Load all 32 A-fragments for this wave's 16x128 tile into registers.
    // Lane L reads row (L&15), byte-aligned contiguous pairs -> global_load_b64.
    v2f afrag[32];
    #pragma unroll
    for (int k = 0; k < 32; ++k)
        afrag[k] = *(const v2f*)&X[(rowbase + col) * DIN + k * 4 + koff];

    v8f acc2 = {};                          // fc2 accumulator: 16x16 output tile

    for (int jt = 0; jt < 16; ++jt) {       // 16 mid-tiles of 16 cols (DMID=256)
        const int j0 = jt * 16;

        // ---- fc1: H_tile = Xtile @ W1[j0:j0+16, :]^T  (32 WMMA k-steps) ----
        v8f acc1 = {};
        #pragma unroll
        for (int k = 0; k < 32; ++k) {
            v2f b = *(const v2f*)&W1[(j0 + col) * DIN + k * 4 + koff];
            acc1 = __builtin_amdgcn_wmma_f32_16x16x4_f32(
                false, afrag[k], false, b, (short)0, acc1, false, false);
        }

        // ---- ReLU, round-trip through LDS to reshape C-layout -> A-layout ----
        #pragma unroll
        for (int r = 0; r < 8; ++r)
            hs[w][r + 8 * hi][col] = fmaxf(acc1[r], 0.0f);
        // Same-wave LDS ops are in-order (ISA 7.3); compiler inserts s_wait_dscnt.

        // ---- fc2: acc2 += relu(H_tile) @ W2[:, j0:j0+16]^T (4 WMMA k-steps) ----
        #pragma unroll
        for (int c = 0; c < 4; ++c) {
            v2f a = *(const v2f*)&hs[w][col][c * 4 + koff];
            v2f b = *(const v2f*)&W2[col * DMID + j0 + c * 4 + koff];
            acc2 = __builtin_amdgcn_wmma_f32_16x16x4_f32(
                false, a, false, b, (short)0, acc2, false, false);
        }
    }

    // Write the 16x16 output tile.
    #pragma unroll
    for (int r = 0; r < 8; ++r)
        out[(rowbase + r + 8 * hi) * DOUT + col] = acc2[r];
}

// ---------------------------------------------------------------------------
// KDE stage: out[b,d] = sum_n w*Y[n,d] / sum_n w,  w = exp(-0.5*((Xw[n,d]-Zw[b,d])/h)^2)
// Block = 16 batch rows x 16 features; Xw/Y chunks staged in LDS and shared
// by all 16 batch rows (16x read amplification out of LDS instead of L2).
// ---------------------------------------------------------------------------
#define CHUNK 256

__global__ __launch_bounds__(256) void kde(const float* __restrict__ Xw,
                                           const float* __restrict__ Zw,
                                           const float* __restrict__ Y,
                                           const float* __restrict__ hptr,
                                           float* __restrict__ out) {
    __shared__ __align__(16) float xs[CHUNK * DOUT];
    __shared__ __align__(16) float ys[CHUNK * DOUT];

    const int tid = threadIdx.x;
    const int d   = tid & 15;
    const int bl  = tid >> 4;
    const int b   = blockIdx.x * 16 + bl;

    const float invh = 1.0f / hptr[0];
    const float z    = Zw[b * DOUT + d];

    float s = 0.0f, t = 0.0f;

    for (int n0 = 0; n0 < NPTS; n0 += CHUNK) {
        __syncthreads();
        // Cooperative chunk load: CHUNK*16 floats per array, float4-coalesced.
        #pragma unroll
        for (int i = 0; i < 4; ++i) {
            int v = i * 256 + tid;          // float4 index (0..1023)
            *(float4*)&xs[v * 4] = *(const float4*)&Xw[n0 * DOUT + v * 4];
            *(float4*)&ys[v * 4] = *(const float4*)&Y[n0 * DOUT + v * 4];
        }
        // Prefetch next chunk into cache while we crunch this one.
        if (n0 + CHUNK < NPTS) {
            __builtin_prefetch(&Xw[(n0 + CHUNK) * DOUT + tid * 16], 0, 0);
            __builtin_prefetch(&Y[(n0 + CHUNK) * DOUT + tid * 16], 0, 0);
        }
        __syncthreads();

        #pragma unroll 8
        for (int n = 0; n < CHUNK; ++n) {
            float diff = (xs[n * DOUT + d] - z) * invh;
            float wgt  = __expf(-0.5f * diff * diff);   // v_exp_f32 path
            s += wgt;
            t = fmaf(wgt, ys[n * DOUT + d], t);
        }
    }

    out[b * DOUT + d] = t / s;
}

// ---------------------------------------------------------------------------
// Inputs (setup_inputs order): x[512,128], calc_X[8192,128], calc_Y[8192,16],
// W1[256,128], W2[16,256], h[1].   Output: [512,16] float32.
// Workspace: Xw (8192*16 f32) + Zw (512*16 f32) = 544 KiB.
// ---------------------------------------------------------------------------
extern "C" void kernel_launch(void* const* d_in, const int* in_sizes, int n_in,
                              void* d_out, int out_size, void* d_ws, size_t ws_size,
                              hipStream_t stream) {
    const float* x   = (const float*)d_in[0];
    const float* cX  = (const float*)d_in[1];
    const float* cY  = (const float*)d_in[2];
    const float* W1  = (const float*)d_in[3];
    const float* W2  = (const float*)d_in[4];
    const float* h   = (const float*)d_in[5];
    float*       out = (float*)d_out;

    float* Xw = (float*)d_ws;               // 8192*16
    float* Zw = Xw + NPTS * DOUT;           // 512*16

    // 16-row tiles: 4 per block (one per wave).
    mlp_wmma<<<NPTS / 64, 128, 0, stream>>>(cX, W1, W2, Xw);   // 512 tiles
    mlp_wmma<<<BQ   / 64, 128, 0, stream>>>(x,  W1, W2, Zw);   // 32 tiles

    kde<<<BQ / 16, 256, 0, stream>>>(Xw, Zw, cY, h, out);
}